// Net_orig_46712064311561
// MI455X (gfx1250) — compile-verified
//
#include <hip/hip_runtime.h>
#include <math.h>

#define N_NODES   100000
#define N_FEAT    256
#define HIDDEN    128
#define N_CLASSES 40

typedef __attribute__((ext_vector_type(2))) float v2f;
typedef __attribute__((ext_vector_type(8))) float v8f;
typedef int v4i __attribute__((vector_size(16)));

// ---- gfx1250 async global->LDS path (ASYNCcnt-tracked), with safe fallback ----
#if defined(__has_builtin)
#if __has_builtin(__builtin_amdgcn_global_load_async_to_lds_b128)
#define ASYNC_LDS 1
#endif
#endif
#ifndef ASYNC_LDS
#define ASYNC_LDS 0
#endif

__device__ __forceinline__ void lds_copy16(const float* g, float* lds) {
#if ASYNC_LDS
    __builtin_amdgcn_global_load_async_to_lds_b128(
        (__attribute__((address_space(1))) v4i*)g,
        (__attribute__((address_space(3))) v4i*)lds,
        0, 0);
#else
    *(float4*)lds = *(const float4*)g;
#endif
}

__device__ __forceinline__ void lds_copy_wait() {
#if ASYNC_LDS
#if __has_builtin(__builtin_amdgcn_s_wait_asynccnt)
    __builtin_amdgcn_s_wait_asynccnt(0);
#else
    asm volatile("s_wait_asynccnt 0x0" ::: "memory");
#endif
#endif
}

// ---------------- degree / normalization ----------------
__global__ void k_deg_init(float* __restrict__ deg) {
    int i = blockIdx.x * blockDim.x + threadIdx.x;
    if (i < N_NODES) deg[i] = 1.0f;             // self-loop contribution
}

__global__ void k_deg_accum(const int* __restrict__ dst, float* __restrict__ deg, int nE) {
    int i = blockIdx.x * blockDim.x + threadIdx.x;
    if (i < nE) atomicAdd(&deg[dst[i]], 1.0f);
}

__global__ void k_dinv(float* __restrict__ deg) {
    int i = blockIdx.x * blockDim.x + threadIdx.x;
    if (i < N_NODES) {
        float d = deg[i];
        deg[i] = (d > 0.0f) ? rsqrtf(d) : 0.0f; // in place: deg -> dinv
    }
}

// per-edge normalization factor, reused by both layers
__global__ void k_norm(const int* __restrict__ src, const int* __restrict__ dst,
                       const float* __restrict__ dinv, float* __restrict__ enorm, int nE) {
    int e = blockIdx.x * blockDim.x + threadIdx.x;
    if (e < nE) enorm[e] = dinv[src[e]] * dinv[dst[e]];
}

// ---------------- GEMM1: [100000,256] @ [256,128] via V_WMMA_F32_16X16X4_F32 ----
// Block = 256 threads = 8 waves; each wave owns a 16-col tile (8*16 = 128 = HIDDEN).
__global__ void __launch_bounds__(256) k_gemm1(const float* __restrict__ X,
                                               const float* __restrict__ W,
                                               float* __restrict__ XW) {
    __shared__ float As[16 * N_FEAT];           // 16 KB
    const int tid = threadIdx.x;
    const int m0  = blockIdx.x * 16;

    // stage 16x256 A tile into LDS in 16B chunks (async-to-LDS when available)
    for (int c = tid; c < (16 * N_FEAT) / 4; c += 256) {
        int idx = c * 4;
        lds_copy16(&X[(m0 + (idx >> 8)) * N_FEAT + (idx & 255)], &As[idx]);
    }
    lds_copy_wait();
    __syncthreads();

    const int wave = tid >> 5;
    const int lane = tid & 31;
    const int half = lane >> 4;      // 0: lanes 0-15, 1: lanes 16-31
    const int lh   = lane & 15;
    const int n0   = wave * 16;

    v8f c = {};
#pragma unroll 4
    for (int k0 = 0; k0 < N_FEAT; k0 += 4) {
        v2f a, b;
        // A 16x4 f32 layout: VGPR j = K = 2*half + j, row = lh
        a[0] = As[lh * N_FEAT + k0 + 2 * half + 0];
        a[1] = As[lh * N_FEAT + k0 + 2 * half + 1];
        // B 4x16 f32 layout: VGPR j = K row 2*half + j, col = lh
        b[0] = W[(k0 + 2 * half + 0) * HIDDEN + n0 + lh];
        b[1] = W[(k0 + 2 * half + 1) * HIDDEN + n0 + lh];
        c = __builtin_amdgcn_wmma_f32_16x16x4_f32(false, a, false, b, (short)0, c, false, false);
    }
    // D 16x16 f32: VGPR r -> row r + 8*half, col lh
#pragma unroll
    for (int r = 0; r < 8; ++r) {
        XW[(m0 + r + 8 * half) * HIDDEN + n0 + lh] = c[r];
    }
}

// ---------------- GEMM2: [100000,128] @ [128,40], N padded 40 -> 48 (3 wave tiles) ----
__global__ void __launch_bounds__(96) k_gemm2(const float* __restrict__ H,
                                              const float* __restrict__ W,
                                              float* __restrict__ O) {
    __shared__ float Hs[16 * HIDDEN];           // 8 KB
    const int tid = threadIdx.x;
    const int m0  = blockIdx.x * 16;

    for (int c = tid; c < (16 * HIDDEN) / 4; c += 96) {
        int idx = c * 4;
        lds_copy16(&H[(m0 + (idx >> 7)) * HIDDEN + (idx & 127)], &Hs[idx]);
    }
    lds_copy_wait();
    __syncthreads();

    const int wave = tid >> 5;                  // 0..2
    const int lane = tid & 31;
    const int half = lane >> 4;
    const int lh   = lane & 15;
    const int col  = wave * 16 + lh;
    // keep EXEC all-1s at the WMMA: clamp the load address, zero via multiply
    const int   ccol  = (col < N_CLASSES) ? col : (N_CLASSES - 1);
    const float cmask = (col < N_CLASSES) ? 1.0f : 0.0f;

    v8f c = {};
#pragma unroll 4
    for (int k0 = 0; k0 < HIDDEN; k0 += 4) {
        v2f a, b;
        a[0] = Hs[lh * HIDDEN + k0 + 2 * half + 0];
        a[1] = Hs[lh * HIDDEN + k0 + 2 * half + 1];
        b[0] = W[(k0 + 2 * half + 0) * N_CLASSES + ccol] * cmask;
        b[1] = W[(k0 + 2 * half + 1) * N_CLASSES + ccol] * cmask;
        c = __builtin_amdgcn_wmma_f32_16x16x4_f32(false, a, false, b, (short)0, c, false, false);
    }
    if (col < N_CLASSES) {
#pragma unroll
        for (int r = 0; r < 8; ++r) {
            O[(m0 + r + 8 * half) * N_CLASSES + col] = c[r];
        }
    }
}

// ---------------- aggregation ----------------
// agg[i,f] = xw[i,f] * dinv[i]^2   (self-loop term, also initializes buffer)
template <int F>
__global__ void k_selfloop(const float* __restrict__ xw, const float* __restrict__ dinv,
                           float* __restrict__ agg) {
    long long t = (long long)blockIdx.x * blockDim.x + threadIdx.x;
    if (t < (long long)N_NODES * F) {
        int i = (int)(t / F);
        float di = dinv[i];
        agg[t] = xw[t] * di * di;
    }
}

// agg[dst,f] += xw[src,f] * enorm[e]
template <int F>
__global__ void k_scatter(const int* __restrict__ src, const int* __restrict__ dst,
                          const float* __restrict__ enorm, const float* __restrict__ xw,
                          float* __restrict__ agg, int nE) {
    long long t = (long long)blockIdx.x * blockDim.x + threadIdx.x;
    if (t < (long long)nE * F) {
        int e = (int)(t / F);
        int f = (int)(t - (long long)e * F);
        int s = src[e];
        int d = dst[e];
        atomicAdd(&agg[(long long)d * F + f], xw[(long long)s * F + f] * enorm[e]);
    }
}

__global__ void k_bias_relu(float* __restrict__ h, const float* __restrict__ b) {
    int t = blockIdx.x * blockDim.x + threadIdx.x;
    if (t < N_NODES * HIDDEN) {
        float v = h[t] + b[t & (HIDDEN - 1)];
        h[t] = (v > 0.0f) ? v : 0.0f;
    }
}

// ---------------- bias + log_softmax (in place on d_out) ----------------
__global__ void k_logsoftmax(float* __restrict__ out, const float* __restrict__ b2) {
    int i = blockIdx.x * blockDim.x + threadIdx.x;
    if (i >= N_NODES) return;
    float v[N_CLASSES];
    float mx = -INFINITY;
#pragma unroll
    for (int c = 0; c < N_CLASSES; ++c) {
        v[c] = out[i * N_CLASSES + c] + b2[c];
        mx = fmaxf(mx, v[c]);
    }
    float s = 0.0f;
#pragma unroll
    for (int c = 0; c < N_CLASSES; ++c) s += expf(v[c] - mx);
    float ls = logf(s);
#pragma unroll
    for (int c = 0; c < N_CLASSES; ++c) out[i * N_CLASSES + c] = v[c] - mx - ls;
}

extern "C" void kernel_launch(void* const* d_in, const int* in_sizes, int n_in,
                              void* d_out, int out_size, void* d_ws, size_t ws_size,
                              hipStream_t stream) {
    const float* x  = (const float*)d_in[0];
    const int*   ei = (const int*)d_in[1];
    const float* W1 = (const float*)d_in[2];
    const float* b1 = (const float*)d_in[3];
    const float* W2 = (const float*)d_in[4];
    const float* b2 = (const float*)d_in[5];
    float*       out = (float*)d_out;

    const int nE  = in_sizes[1] / 2;      // 1,600,000
    const int* src = ei;                  // edge_index[0]
    const int* dst = ei + nE;             // edge_index[1]

    // workspace layout (floats, 256B-ish aligned chunks)
    float* ws    = (float*)d_ws;
    float* dinv  = ws;                                        // [N_NODES]
    float* enorm = dinv  + (((size_t)N_NODES + 63) & ~(size_t)63);   // [nE]
    float* xw1   = enorm + (((size_t)nE + 63) & ~(size_t)63); // [N_NODES, HIDDEN]
    float* hbuf  = xw1   + (size_t)N_NODES * HIDDEN;          // [N_NODES, HIDDEN]
    float* olin  = xw1;                                       // reuse xw1 after ReLU

    const int B = 256;
    // 1) degrees -> dinv -> per-edge norm
    k_deg_init<<<(N_NODES + B - 1) / B, B, 0, stream>>>(dinv);
    k_deg_accum<<<(nE + B - 1) / B, B, 0, stream>>>(dst, dinv, nE);
    k_dinv<<<(N_NODES + B - 1) / B, B, 0, stream>>>(dinv);
    k_norm<<<(nE + B - 1) / B, B, 0, stream>>>(src, dst, dinv, enorm, nE);

    // 2) GEMM1 (WMMA f32)
    k_gemm1<<<N_NODES / 16, 256, 0, stream>>>(x, W1, xw1);

    // 3) layer-1 aggregation + bias + relu  -> hbuf
    {
        long long tn = (long long)N_NODES * HIDDEN;
        k_selfloop<HIDDEN><<<(unsigned)((tn + B - 1) / B), B, 0, stream>>>(xw1, dinv, hbuf);
        long long te = (long long)nE * HIDDEN;
        k_scatter<HIDDEN><<<(unsigned)((te + B - 1) / B), B, 0, stream>>>(src, dst, enorm, xw1, hbuf, nE);
        k_bias_relu<<<(unsigned)((tn + B - 1) / B), B, 0, stream>>>(hbuf, b1);
    }

    // 4) GEMM2 (WMMA f32) -> olin (reuses xw1 region)
    k_gemm2<<<N_NODES / 16, 96, 0, stream>>>(hbuf, W2, olin);

    // 5) layer-2 aggregation -> d_out, then bias + log_softmax in place
    {
        long long tn = (long long)N_NODES * N_CLASSES;
        k_selfloop<N_CLASSES><<<(unsigned)((tn + B - 1) / B), B, 0, stream>>>(olin, dinv, out);
        long long te = (long long)nE * N_CLASSES;
        k_scatter<N_CLASSES><<<(unsigned)((te + B - 1) / B), B, 0, stream>>>(src, dst, enorm, olin, out, nE);
        k_logsoftmax<<<(N_NODES + B - 1) / B, B, 0, stream>>>(out, b2);
    }
}